// GCNNet_28991029248691
// MI455X (gfx1250) — compile-verified
//
#include <hip/hip_runtime.h>
#include <math.h>

typedef __attribute__((ext_vector_type(2))) float v2f;
typedef __attribute__((ext_vector_type(8))) float v8f;

// ---------------- small elementwise kernels ----------------

__global__ void fold_gate_w1(const float* __restrict__ W1, const float* __restrict__ xw,
                             float* __restrict__ Wg, int K, int D) {
  int idx = blockIdx.x * blockDim.x + threadIdx.x;
  if (idx >= K * D) return;
  int k = idx / D;
  float w = xw[k];
  float gate = (fabsf(w) > 0.0f) ? (1.0f / (1.0f + expf(-w))) : 0.0f;
  Wg[idx] = W1[idx] * gate;
}

__global__ void init_const(float* __restrict__ p, float v, int n) {
  int i = blockIdx.x * blockDim.x + threadIdx.x;
  if (i < n) p[i] = v;
}

__global__ void deg_accum(const int* __restrict__ col, const float* __restrict__ ew,
                          float* __restrict__ deg, int E) {
  int e = blockIdx.x * blockDim.x + threadIdx.x;
  if (e < E) atomicAdd(&deg[col[e]], ew[e]);
}

__global__ void finalize_dinv(float* __restrict__ deg, int n) {
  int i = blockIdx.x * blockDim.x + threadIdx.x;
  if (i >= n) return;
  float d = deg[i];
  deg[i] = (d > 0.0f) ? rsqrtf(d) : 0.0f;
}

// ---------------- fp32 WMMA GEMM: C[M,N] = A[M,K] @ B[K,N] ----------------
// One wave computes a 16 x (16*NT) output strip. N (=NN) is a compile-time
// stride so all B/C addressing folds into immediate offsets; K stepped by 8
// with batched B loads so the scheduler can clause loads and issue 2*NT
// back-to-back WMMAs per chunk.
// V_WMMA_F32_16X16X4_F32 fragment layouts (ISA 7.12.2):
//   A: lanes 0-15 M=lane, VGPR0/1 = K+0/K+1 ; lanes 16-31 = K+2/K+3.
//   B: lanes index N; VGPR0/1 = K rows (k+2*half, +1).
//   C/D: VGPR i -> row m0+i (lanes 0-15) / m0+8+i (lanes 16-31).
template <int NT, int NN>
__global__ __launch_bounds__(32)
void gemm_wmma_f32_t(const float* __restrict__ A, const float* __restrict__ B,
                     float* __restrict__ C, int M, int K) {
  const int lane = threadIdx.x & 31;
  const int half = lane >> 4;   // 0 or 1
  const int l16  = lane & 15;
  const int m0 = blockIdx.x * 16;
  const int n0 = blockIdx.y * (16 * NT);

  const v8f vzero = {0.f, 0.f, 0.f, 0.f, 0.f, 0.f, 0.f, 0.f};
  v8f acc[NT];
#pragma unroll
  for (int t = 0; t < NT; ++t) acc[t] = vzero;

  const float* __restrict__ arow = A + (size_t)(m0 + l16) * K + 2 * half;
  const float* __restrict__ bp   = B + (size_t)(2 * half) * NN + n0 + l16;

  for (int k = 0; k < K; k += 8) {
    v2f a0, a1;
    a0.x = arow[0];
    a0.y = arow[1];
    a1.x = arow[4];
    a1.y = arow[5];
    v2f b0[NT], b1[NT];
#pragma unroll
    for (int t = 0; t < NT; ++t) {
      const float* __restrict__ p = bp + t * 16;
      b0[t].x = p[0];
      b0[t].y = p[NN];
      b1[t].x = p[4 * NN];
      b1[t].y = p[5 * NN];
    }
#pragma unroll
    for (int t = 0; t < NT; ++t)
      acc[t] = __builtin_amdgcn_wmma_f32_16x16x4_f32(false, a0, false, b0[t],
                                                     (short)0, acc[t], false, false);
#pragma unroll
    for (int t = 0; t < NT; ++t)
      acc[t] = __builtin_amdgcn_wmma_f32_16x16x4_f32(false, a1, false, b1[t],
                                                     (short)0, acc[t], false, false);
    arow += 8;
    bp += 8 * NN;
  }

#pragma unroll
  for (int t = 0; t < NT; ++t) {
#pragma unroll
    for (int i = 0; i < 8; ++i) {
      const int m = m0 + i + 8 * half;
      C[(size_t)m * NN + (n0 + t * 16 + l16)] = acc[t][i];
    }
  }
}

// generic fallback (runtime N, NT=1, K multiple of 4)
__global__ __launch_bounds__(32)
void gemm_wmma_f32_gen(const float* __restrict__ A, const float* __restrict__ B,
                       float* __restrict__ C, int M, int N, int K) {
  const int lane = threadIdx.x & 31;
  const int half = lane >> 4;
  const int l16  = lane & 15;
  const int m0 = blockIdx.x * 16;
  const int n0 = blockIdx.y * 16;

  v8f acc = {0.f, 0.f, 0.f, 0.f, 0.f, 0.f, 0.f, 0.f};
  const float* __restrict__ arow = A + (size_t)(m0 + l16) * K;
  for (int k = 0; k < K; k += 4) {
    const int ka = k + 2 * half;
    v2f a, b;
    a.x = arow[ka + 0];
    a.y = arow[ka + 1];
    const float* __restrict__ p = B + (size_t)ka * N + (n0 + l16);
    b.x = p[0];
    b.y = p[N];
    acc = __builtin_amdgcn_wmma_f32_16x16x4_f32(false, a, false, b, (short)0, acc,
                                                false, false);
  }
#pragma unroll
  for (int i = 0; i < 8; ++i)
    C[(size_t)(m0 + i + 8 * half) * N + (n0 + l16)] = acc[i];
}

// ---------------- edge scatter-add ----------------
// out[col[e], f] += src[row[e], f] * dinv[row[e]] * ew[e] * dinv[col[e]]
// Vectorized: one thread handles 4 consecutive features (b128 gather + 4 atomics).
// LOGQ = log2(F/4).

template <int LOGQ>
__global__ void edge_aggregate_v4(const float* __restrict__ src,
                                  const int* __restrict__ row, const int* __restrict__ col,
                                  const float* __restrict__ ew, const float* __restrict__ dinv,
                                  float* __restrict__ out, int E) {
  long long idx = (long long)blockIdx.x * blockDim.x + threadIdx.x;
  long long tot = (long long)E << LOGQ;
  if (idx >= tot) return;
  int e = (int)(idx >> LOGQ);
  int q = (int)(idx & ((1 << LOGQ) - 1));
  int r = row[e], c = col[e];
  float w = dinv[r] * ew[e] * dinv[c];
  const float4 v = *(const float4*)(src + ((((size_t)r << LOGQ) + q) << 2));
  float* o = out + ((((size_t)c << LOGQ) + q) << 2);
  atomicAdd(o + 0, v.x * w);
  atomicAdd(o + 1, v.y * w);
  atomicAdd(o + 2, v.z * w);
  atomicAdd(o + 3, v.w * w);
}

__global__ void edge_aggregate_gen(const float* __restrict__ src,
                                   const int* __restrict__ row, const int* __restrict__ col,
                                   const float* __restrict__ ew, const float* __restrict__ dinv,
                                   float* __restrict__ out, int E, int F) {
  long long idx = (long long)blockIdx.x * blockDim.x + threadIdx.x;
  long long tot = (long long)E * F;
  if (idx >= tot) return;
  int e = (int)(idx / F);
  int f = (int)(idx - (long long)e * F);
  int r = row[e], c = col[e];
  float w = dinv[r] * ew[e] * dinv[c];
  atomicAdd(&out[(size_t)c * F + f], src[(size_t)r * F + f] * w);
}

// ---------------- layer-1 epilogue: self-loop + bias + relu (in place) ----------------

__global__ void selfloop_bias_relu(float* __restrict__ agg, const float* __restrict__ xw,
                                   const float* __restrict__ dinv, const float* __restrict__ b,
                                   int n, int F) {
  long long idx = (long long)blockIdx.x * blockDim.x + threadIdx.x;
  if (idx >= (long long)n * F) return;
  int i = (int)(idx / F);
  int f = (int)(idx - (long long)i * F);
  float d = dinv[i];
  float v = agg[idx] + xw[idx] * d * d + b[f];
  agg[idx] = fmaxf(v, 0.0f);
}

// ---------------- layer-2 epilogue + log_softmax ----------------

__global__ void final_logsoftmax(const float* __restrict__ agg2, const float* __restrict__ t,
                                 const float* __restrict__ dinv, const float* __restrict__ b2,
                                 float* __restrict__ out, int n, int C) {
  int i = blockIdx.x * blockDim.x + threadIdx.x;
  if (i >= n) return;
  float d = dinv[i];
  float d2 = d * d;
  float vals[64];
  float mx = -INFINITY;
  const size_t base = (size_t)i * C;
  for (int f = 0; f < C; ++f) {
    float v = agg2[base + f] + t[base + f] * d2 + b2[f];
    vals[f] = v;
    mx = fmaxf(mx, v);
  }
  float s = 0.0f;
  for (int f = 0; f < C; ++f) s += expf(vals[f] - mx);
  float lse = mx + logf(s);
  for (int f = 0; f < C; ++f) out[base + f] = vals[f] - lse;
}

// ---------------- host ----------------

static inline unsigned cdiv_ll(long long a, long long b) { return (unsigned)((a + b - 1) / b); }

extern "C" void kernel_launch(void* const* d_in, const int* in_sizes, int n_in,
                              void* d_out, int out_size, void* d_ws, size_t ws_size,
                              hipStream_t stream) {
  const float* x       = (const float*)d_in[0];
  const int*   ei      = (const int*)d_in[1];
  const float* ea      = (const float*)d_in[2];
  const float* xweight = (const float*)d_in[3];
  const float* W1      = (const float*)d_in[4];
  const float* b1      = (const float*)d_in[5];
  const float* W2      = (const float*)d_in[6];
  const float* b2      = (const float*)d_in[7];

  const int E  = in_sizes[2];
  const int F  = in_sizes[3];        // 512
  const int D  = in_sizes[5];        // 256
  const int Cn = in_sizes[7];        // 64
  const int N  = in_sizes[0] / F;    // 50000

  const int* row = ei;
  const int* col = ei + E;

  // workspace carve-up (floats)
  float* ws = (float*)d_ws;
  float* Wg1  = ws;
  size_t off  = (size_t)F * D;
  float* dinv = ws + off;  off += (size_t)N;
  float* xw1  = ws + off;  off += (size_t)N * D;
  float* agg1 = ws + off;  off += (size_t)N * D;
  float* t2   = xw1;                       // reuse: xw1 dead after layer-1 epilogue
  float* agg2 = xw1 + (size_t)N * Cn;      // 2*N*Cn <= N*D
  (void)ws_size; (void)n_in; (void)out_size;

  // 1. fold sparse-feature gate into W1
  {
    int tot = F * D;
    fold_gate_w1<<<(tot + 255) / 256, 256, 0, stream>>>(W1, xweight, Wg1, F, D);
  }

  // 2. degrees (self-loop weight 1) -> dinv = rsqrt(deg)
  init_const<<<(N + 255) / 256, 256, 0, stream>>>(dinv, 1.0f, N);
  deg_accum<<<(E + 255) / 256, 256, 0, stream>>>(col, ea, dinv, E);
  finalize_dinv<<<(N + 255) / 256, 256, 0, stream>>>(dinv, N);

  // 3. xw1 = x @ Wg1   (M=N, K=F, N=D)
  if ((N % 16) == 0 && (F % 8) == 0 && D == 256) {
    gemm_wmma_f32_t<4, 256><<<dim3(N / 16, 256 / 64), 32, 0, stream>>>(x, Wg1, xw1, N, F);
  } else {
    gemm_wmma_f32_gen<<<dim3((N + 15) / 16, (D + 15) / 16), 32, 0, stream>>>(x, Wg1, xw1, N, D, F);
  }

  // 4. agg1 = 0 ; scatter-add edges at F=D
  {
    int tot = (int)((size_t)N * D);
    init_const<<<(tot + 255) / 256, 256, 0, stream>>>(agg1, 0.0f, tot);
  }
  if (D == 256) {
    edge_aggregate_v4<6><<<cdiv_ll((long long)E << 6, 256), 256, 0, stream>>>(
        xw1, row, col, ea, dinv, agg1, E);
  } else if (D == 64) {
    edge_aggregate_v4<4><<<cdiv_ll((long long)E << 4, 256), 256, 0, stream>>>(
        xw1, row, col, ea, dinv, agg1, E);
  } else {
    edge_aggregate_gen<<<cdiv_ll((long long)E * D, 256), 256, 0, stream>>>(
        xw1, row, col, ea, dinv, agg1, E, D);
  }

  // 5. h = relu(agg1 + xw1*dinv^2 + b1)   (in place in agg1)
  selfloop_bias_relu<<<cdiv_ll((long long)N * D, 256), 256, 0, stream>>>(
      agg1, xw1, dinv, b1, N, D);

  // 6. t2 = h @ W2   (M=N, K=D, N=Cn)  -- reuses xw1 storage
  if ((N % 16) == 0 && (D % 8) == 0 && Cn == 64) {
    gemm_wmma_f32_t<4, 64><<<dim3(N / 16, 1), 32, 0, stream>>>(agg1, W2, t2, N, D);
  } else {
    gemm_wmma_f32_gen<<<dim3((N + 15) / 16, (Cn + 15) / 16), 32, 0, stream>>>(agg1, W2, t2, N, Cn, D);
  }

  // 7. agg2 = 0 ; scatter-add edges at F=Cn
  {
    int tot = (int)((size_t)N * Cn);
    init_const<<<(tot + 255) / 256, 256, 0, stream>>>(agg2, 0.0f, tot);
  }
  if (Cn == 64) {
    edge_aggregate_v4<4><<<cdiv_ll((long long)E << 4, 256), 256, 0, stream>>>(
        t2, row, col, ea, dinv, agg2, E);
  } else {
    edge_aggregate_gen<<<cdiv_ll((long long)E * Cn, 256), 256, 0, stream>>>(
        t2, row, col, ea, dinv, agg2, E, Cn);
  }

  // 8. out = log_softmax(agg2 + t2*dinv^2 + b2)
  final_logsoftmax<<<(N + 127) / 128, 128, 0, stream>>>(
      agg2, t2, dinv, b2, (float*)d_out, N, Cn);
}